// GlobalAggregator_34144990003659
// MI455X (gfx1250) — compile-verified
//
#include <hip/hip_runtime.h>

typedef __attribute__((ext_vector_type(2))) float v2f;
typedef __attribute__((ext_vector_type(8))) float v8f;

#define NI     5      // n rows per block (50 % 5 == 0)
#define KADJ   12
#define DD     128
#define NROW   50

// LDS layout (float offsets)
#define A_STRIDE  132
#define G_STRIDE  260
#define OFF_ADJ   0                                // [NI][12][128] = 7680
#define OFF_A     (OFF_ADJ + NI*KADJ*DD)           // [80][132]     = 10560
#define OFF_G     (OFF_A + 80*A_STRIDE)            // [16][260]     = 4160
#define OFF_S     (OFF_G + 16*G_STRIDE)            // [NI*16]       = 80
#define OFF_ALPHA (OFF_S + NI*16)                  // [NI*12]       = 60
#define SMEM_FLOATS (OFF_ALPHA + NI*KADJ)

#if defined(__has_builtin)
#if __has_builtin(__builtin_amdgcn_global_load_async_to_lds_b128) && \
    __has_builtin(__builtin_amdgcn_s_wait_asynccnt)
#define USE_ASYNC_LDS 1
#endif
#endif

// Pointer types matching the builtin's signature:
//   (v4i addrspace(1)* src, v4i addrspace(3)* dst, imm offset, imm cpol)
typedef int v4i_b128 __attribute__((vector_size(16)));
typedef __attribute__((address_space(1))) v4i_b128 g_v4i;
typedef __attribute__((address_space(3))) v4i_b128 l_v4i;

__global__ __launch_bounds__(256)
void ga_kernel(const float* __restrict__ itm,
               const float* __restrict__ ave,
               const float* __restrict__ adj,
               const float* __restrict__ wgt,
               const float* __restrict__ w1,
               const float* __restrict__ w2,
               const float* __restrict__ w3,
               float* __restrict__ out)
{
    extern __shared__ float sm[];

    const int tid  = threadIdx.x;
    const int lane = tid & 31;
    const int wv   = tid >> 5;          // wave 0..7 -> 16-col N tile
    const int half = lane >> 4;         // 0 or 1
    const int l16  = lane & 15;
    const int b    = blockIdx.y;
    const int n0   = blockIdx.x * NI;
    const int bn   = b * NROW + n0;

    const float* adjB = adj + (size_t)bn * (KADJ * DD);

    // ---------------- Phase 1a: stage adj -> LDS (async DMA if available),
    //                  overlap with independent G-matrix / score init -------
#if USE_ASYNC_LDS
    for (int v = tid; v < (NI * KADJ * DD) / 4; v += 256) {
        g_v4i* gp = (g_v4i*)(uintptr_t)(adjB + 4 * v);
        l_v4i* lp = (l_v4i*)(unsigned)(uintptr_t)(&sm[OFF_ADJ + 4 * v]);
        __builtin_amdgcn_global_load_async_to_lds_b128(gp, lp, 0, 0);
    }
#else
    for (int i = tid; i < NI * KADJ * DD; i += 256)
        sm[OFF_ADJ + i] = adjB[i];
#endif

    // G matrix: rows 0..4 cols 0..127 = itm ; rows 5..15 zero
    for (int i = tid; i < 16 * 256; i += 256) {
        int r = i >> 8, c = i & 255;
        if (r >= NI)
            sm[OFF_G + r * G_STRIDE + c] = 0.0f;
        else if (c < DD)
            sm[OFF_G + r * G_STRIDE + c] = itm[(size_t)(bn + r) * DD + c];
    }
    if (tid < NI * 16) sm[OFF_S + tid] = 0.0f;

#if USE_ASYNC_LDS
    __builtin_amdgcn_s_wait_asynccnt(0);
#endif
    __syncthreads();

    // ---------------- Phase 1b: build padded A from LDS adj ---------------
    for (int i = tid; i < 80 * A_STRIDE; i += 256) {
        int r  = i / A_STRIDE;
        int c  = i - r * A_STRIDE;
        int ni = r >> 4;
        int rr = r & 15;
        float v = 0.0f;
        if (rr < KADJ) {
            if (c < DD)
                v = ave[(size_t)(bn + ni) * DD + c] *
                    sm[OFF_ADJ + (ni * KADJ + rr) * DD + c];
            else if (c == DD)
                v = wgt[(bn + ni) * KADJ + rr];
        }
        sm[OFF_A + i] = v;
    }
    __syncthreads();

    // ---------------- Phase 2: GEMM1 (A[16x132] x w1[132x16]) + scores ----
    {
        const int   ncol = wv * 16 + l16;
        const float w2n  = w2[ncol];

        for (int ni = 0; ni < NI; ++ni) {
            const float* arow = &sm[OFF_A + (ni * 16 + l16) * A_STRIDE];
            v8f acc = {0.f, 0.f, 0.f, 0.f, 0.f, 0.f, 0.f, 0.f};

            for (int kk = 0; kk < 32; ++kk) {
                int k = kk * 4 + half * 2;
                v2f a;  a.x = arow[k];  a.y = arow[k + 1];
                v2f bb; const float* bp = w1 + (size_t)k * DD + ncol;
                bb.x = bp[0]; bb.y = bp[DD];
                acc = __builtin_amdgcn_wmma_f32_16x16x4_f32(
                          false, a, false, bb, (short)0, acc, false, false);
            }
            {   // k = 128..131 : only row 128 of w1 exists (concat column)
                int k = 128 + half * 2;
                v2f a;  a.x = arow[k];  a.y = arow[k + 1];   // LDS zero-padded
                v2f bb;
                bb.x = (half == 0) ? w1[(size_t)128 * DD + ncol] : 0.0f;
                bb.y = 0.0f;
                acc = __builtin_amdgcn_wmma_f32_16x16x4_f32(
                          false, a, false, bb, (short)0, acc, false, false);
            }

            // leaky-relu, dot with w2 column slice, reduce 16-lane halves
            #pragma unroll
            for (int i = 0; i < 8; ++i) {
                float h = acc[i];
                h = (h > 0.0f) ? h : 0.2f * h;
                float p = h * w2n;
                p += __shfl_xor(p, 8);
                p += __shfl_xor(p, 4);
                p += __shfl_xor(p, 2);
                p += __shfl_xor(p, 1);
                if (l16 == 0)
                    atomicAdd(&sm[OFF_S + ni * 16 + i + 8 * half], p);
            }
        }
    }
    __syncthreads();

    // ---------------- Phase 3: softmax over K=12 (one thread per ni) ------
    if (tid < NI) {
        float mx = -3.0e38f;
        for (int k = 0; k < KADJ; ++k)
            mx = fmaxf(mx, sm[OFF_S + tid * 16 + k]);
        float ssum = 0.0f;
        for (int k = 0; k < KADJ; ++k) {
            float e = __expf(sm[OFF_S + tid * 16 + k] - mx);
            sm[OFF_ALPHA + tid * KADJ + k] = e;
            ssum += e;
        }
        float inv = 1.0f / ssum;
        for (int k = 0; k < KADJ; ++k)
            sm[OFF_ALPHA + tid * KADJ + k] *= inv;
    }
    __syncthreads();

    // ---------------- Phase 4: att = sum_k alpha_k * adj_k  -> G[:,128:] --
    for (int i = tid; i < NI * DD; i += 256) {
        int ni = i >> 7, d = i & (DD - 1);
        float a2 = 0.0f;
        #pragma unroll
        for (int k = 0; k < KADJ; ++k)
            a2 += sm[OFF_ALPHA + ni * KADJ + k] *
                  sm[OFF_ADJ + (ni * KADJ + k) * DD + d];
        sm[OFF_G + ni * G_STRIDE + DD + d] = a2;
    }
    __syncthreads();

    // ---------------- Phase 5: GEMM2 (G[16x256] x w3[256x16]) + relu ------
    {
        const int ncol = wv * 16 + l16;
        v8f acc = {0.f, 0.f, 0.f, 0.f, 0.f, 0.f, 0.f, 0.f};

        for (int kk = 0; kk < 64; ++kk) {
            int k = kk * 4 + half * 2;
            v2f a;
            a.x = sm[OFF_G + l16 * G_STRIDE + k];
            a.y = sm[OFF_G + l16 * G_STRIDE + k + 1];
            v2f bb; const float* bp = w3 + (size_t)k * DD + ncol;
            bb.x = bp[0]; bb.y = bp[DD];
            acc = __builtin_amdgcn_wmma_f32_16x16x4_f32(
                      false, a, false, bb, (short)0, acc, false, false);
        }

        // lanes 0..15 hold rows M=0..7 in VGPRs 0..7; rows 0..4 are real n's
        if (half == 0) {
            #pragma unroll
            for (int i = 0; i < NI; ++i) {
                float v = acc[i];
                out[(size_t)(bn + i) * DD + ncol] = fmaxf(v, 0.0f);
            }
        }
    }
}

extern "C" void kernel_launch(void* const* d_in, const int* in_sizes, int n_in,
                              void* d_out, int out_size, void* d_ws, size_t ws_size,
                              hipStream_t stream) {
    (void)n_in; (void)out_size; (void)d_ws; (void)ws_size;
    const float* itm = (const float*)d_in[1];
    const float* ave = (const float*)d_in[2];
    const float* adj = (const float*)d_in[3];
    const float* wgt = (const float*)d_in[4];
    const float* w1  = (const float*)d_in[5];
    const float* w2  = (const float*)d_in[6];
    const float* w3  = (const float*)d_in[7];

    const int B = in_sizes[1] / (NROW * DD);   // itm_emb = [B, 50, 128]

    dim3 grid(NROW / NI, B);                   // (10, 256)
    dim3 block(256);                           // 8 wave32 waves
    size_t smBytes = (size_t)SMEM_FLOATS * sizeof(float);  // ~90 KB
    ga_kernel<<<grid, block, smBytes, stream>>>(itm, ave, adj, wgt,
                                                w1, w2, w3, (float*)d_out);
}